// AgreementRouting_34359738711
// MI455X (gfx1250) — compile-verified
//
#include <hip/hip_runtime.h>

typedef float v2f __attribute__((ext_vector_type(2)));
typedef float v8f __attribute__((ext_vector_type(8)));

#define IC      1152
#define OC      10
#define DIMV    16
#define ODIM    (OC * DIMV)        // 160
#define NTHREADS 320               // 10 wave32
#define NWAVES  (NTHREADS / 32)
#define NTILES  (IC / 16)          // 72 row-tiles of 16 i's
#define KCHUNKS (ODIM / 4)         // 40 K-chunks of 4 for wmma f32 16x16x4
#define BM_STRIDE 164              // padded row stride -> conflict-free b64 LDS reads

// LDS float offsets — small hot arrays FIRST so DS 16-bit offset immediates fold
#define L_BM 0                     // expanded block-diag B matrix [16 * BM_STRIDE] = 2624
#define L_V  (L_BM + 16 * BM_STRIDE)   // v [ODIM]
#define L_S  (L_V + ODIM)          // s partials [2*ODIM]
#define L_B  (L_S + 2 * ODIM)      // running logits b_batch [IC*OC] = 11520
#define L_C  (L_B + IC * OC)       // coupling coeffs c      [IC*OC] = 11520
#define L_TOTAL (L_C + IC * OC)    // 26144 floats = 104576 bytes

__global__ __launch_bounds__(NTHREADS)
void AgreementRouting_kernel(const float* __restrict__ u,   // [B, IC, OC, D]
                             const float* __restrict__ bin, // [IC, OC]
                             const int* __restrict__ nit,
                             float* __restrict__ out)       // [B, OC, D]
{
    extern __shared__ float smem[];
    float* lds_bm = smem + L_BM;
    float* lds_v  = smem + L_V;
    float* lds_s  = smem + L_S;
    float* lds_b  = smem + L_B;
    float* lds_c  = smem + L_C;

    const int tid   = threadIdx.x;
    const int bidx  = blockIdx.x;
    const int wave  = tid >> 5;
    const int lane  = tid & 31;
    const int n     = lane & 15;   // WMMA row (A.M) / col (B.N / D.N) index
    const int khalf = lane >> 4;   // which K pair this lane holds

    const float* ubatch = u + (size_t)bidx * IC * ODIM;
    const int niter = nit[0];

    // init running logits = broadcast of input b
    for (int idx = tid; idx < IC * OC; idx += NTHREADS)
        lds_b[idx] = bin[idx];
    __syncthreads();

    for (int t = 0; t <= niter; ++t) {
        // ---------------- Phase A: agreement via WMMA f32 16x16x4 ----------
        // a[i,n] = sum_k u[b,i,k] * Bm[k,n], Bm[k,n] = (k>>4==n) ? v[k] : 0.
        if (t > 0) {
            // materialize Bm rows (per output column n) in LDS, padded stride
            for (int idx = tid; idx < 16 * ODIM; idx += NTHREADS) {
                const int nn = idx / ODIM;
                const int k  = idx - nn * ODIM;
                lds_bm[nn * BM_STRIDE + k] = ((k >> 4) == nn) ? lds_v[k] : 0.f;
            }
            __syncthreads();

            const float* brow = lds_bm + n * BM_STRIDE + 2 * khalf;
            for (int tile = wave; tile < NTILES; tile += NWAVES) {
                const int i0 = tile * 16;
                const float* arow = ubatch + (size_t)(i0 + n) * ODIM + 2 * khalf;
                v8f acc = {0.f, 0.f, 0.f, 0.f, 0.f, 0.f, 0.f, 0.f};
#pragma unroll 8
                for (int kc = 0; kc < KCHUNKS; ++kc) {
                    v2f afrag = *(const v2f*)(arow + kc * 4);
                    v2f bfrag = *(const v2f*)(brow + kc * 4);
                    acc = __builtin_amdgcn_wmma_f32_16x16x4_f32(
                        false, afrag, false, bfrag, (short)0, acc, false, false);
                }
                // D layout: VGPR j, lanes 0-15 -> M=j, lanes 16-31 -> M=j+8; N = lane&15
                if (n < OC) {
#pragma unroll
                    for (int j = 0; j < 8; ++j) {
                        const int i = i0 + j + 8 * khalf;
                        lds_b[i * OC + n] += acc[j];
                    }
                }
            }
            __syncthreads();
        }

        // ---------------- Phase B: softmax over o per input capsule --------
        for (int i = tid; i < IC; i += NTHREADS) {
            float x[OC];
            float m = -3.0e38f;
#pragma unroll
            for (int o = 0; o < OC; ++o) { x[o] = lds_b[i * OC + o]; m = fmaxf(m, x[o]); }
            float sum = 0.f;
#pragma unroll
            for (int o = 0; o < OC; ++o) { x[o] = __expf(x[o] - m); sum += x[o]; }
            const float r = 1.f / sum;
#pragma unroll
            for (int o = 0; o < OC; ++o) lds_c[i * OC + o] = x[o] * r;
        }
        __syncthreads();

        // ---------------- Phase C: s[o,d] = sum_i c[i,o]*u[i,o,d]; squash --
        {
            const int g  = tid / ODIM;        // i-range half (0 or 1)
            const int od = tid % ODIM;        // this thread's (o,d) column
            const int oo = od >> 4;
            const float* ucol = ubatch + od;
            const int ibeg = g * (IC / 2), iend = ibeg + (IC / 2);
            float part = 0.f;
            for (int i = ibeg; i < iend; ++i) {
                __builtin_prefetch(ucol + (size_t)(i + 40) * ODIM, 0, 1);
                part = fmaf(lds_c[i * OC + oo], ucol[(size_t)i * ODIM], part);
            }
            lds_s[g * ODIM + od] = part;
        }
        __syncthreads();
        if (tid < ODIM) lds_s[tid] += lds_s[ODIM + tid];
        __syncthreads();
        if (tid < ODIM) {
            const int oo = tid >> 4;
            float n2 = 0.f;
#pragma unroll
            for (int d = 0; d < DIMV; ++d) {
                const float sv = lds_s[oo * DIMV + d];
                n2 = fmaf(sv, sv, n2);
            }
            const float s     = lds_s[tid];
            const float scale = (n2 / (1.f + n2)) * rsqrtf(n2);
            lds_v[tid] = s * scale;
        }
        __syncthreads();
    }

    if (tid < ODIM)
        out[(size_t)bidx * ODIM + tid] = lds_v[tid];
}

extern "C" void kernel_launch(void* const* d_in, const int* in_sizes, int n_in,
                              void* d_out, int out_size, void* d_ws, size_t ws_size,
                              hipStream_t stream) {
    (void)n_in; (void)out_size; (void)d_ws; (void)ws_size;
    const float* u   = (const float*)d_in[0];
    const float* bin = (const float*)d_in[1];
    const int*   nit = (const int*)d_in[2];
    float* out = (float*)d_out;

    const int batch = in_sizes[0] / (IC * OC * DIMV);   // 256
    const size_t smem_bytes = (size_t)L_TOTAL * sizeof(float);  // ~102 KB dynamic LDS

    AgreementRouting_kernel<<<batch, NTHREADS, smem_bytes, stream>>>(u, bin, nit, out);
}